// GEN16_3496103379562
// MI455X (gfx1250) — compile-verified
//
#include <hip/hip_runtime.h>

// GENConv x16 for MI455X (gfx1250, wave32).
// Edge phase: 3 bandwidth-bound gather/scatter passes (L2-resident node state).
// GEMM phase: V_WMMA_F32_16X16X4_F32 tiles (16x16 out per wave, K-step 4),
// with residual / BatchNorm / ReLU / bias fused into prologue & epilogue.

#define HID  32
#define HID2 64

typedef __attribute__((ext_vector_type(2))) float v2f;
typedef __attribute__((ext_vector_type(8))) float v8f;

__device__ __forceinline__ v8f wmma4(v2f a, v2f b, v8f c) {
  // 8 args: (neg_a, A, neg_b, B, c_mod, C, reuse_a, reuse_b)
  return __builtin_amdgcn_wmma_f32_16x16x4_f32(false, a, false, b, (short)0, c,
                                               false, false);
}

// ---------------------------------------------------------------- utilities
__global__ void k_zero(float* __restrict__ p, long n) {
  long i = (long)blockIdx.x * blockDim.x + threadIdx.x;
  long stride = (long)gridDim.x * blockDim.x;
  for (; i < n; i += stride) p[i] = 0.f;
}

// ---------------------------------------------------------------- edge phase
// one wave per edge, lane = channel. h rows are 128B -> coalesced.
__global__ void k_edge_max(const float* __restrict__ h,
                           const long long* __restrict__ ei,
                           float* __restrict__ mx, long E) {
  long idx = (long)blockIdx.x * blockDim.x + threadIdx.x;
  long e = idx >> 5;
  int c = (int)(idx & 31);
  if (e >= E) return;
  int s = (int)ei[e];
  int d = (int)ei[E + e];
  float m = fmaxf(h[(long)s * HID + c], 0.f) + 1e-7f;  // m > 0 always
  // m > 0 => IEEE bits are monotone as unsigned; mx initialized to 0.0f
  atomicMax((unsigned int*)(mx + (long)d * HID + c), __float_as_uint(m));
}

__global__ void k_edge_expsum(const float* __restrict__ h,
                              const long long* __restrict__ ei,
                              const float* __restrict__ mx,
                              float* __restrict__ den, long E) {
  long idx = (long)blockIdx.x * blockDim.x + threadIdx.x;
  long e = idx >> 5;
  int c = (int)(idx & 31);
  if (e >= E) return;
  int s = (int)ei[e];
  int d = (int)ei[E + e];
  float m = fmaxf(h[(long)s * HID + c], 0.f) + 1e-7f;
  float ex = __expf(m - mx[(long)d * HID + c]);
  atomicAdd(den + (long)d * HID + c, ex);
}

__global__ void k_edge_agg(const float* __restrict__ h,
                           const long long* __restrict__ ei,
                           const float* __restrict__ mx,
                           const float* __restrict__ den,
                           float* __restrict__ agg, long E) {
  long idx = (long)blockIdx.x * blockDim.x + threadIdx.x;
  long e = idx >> 5;
  int c = (int)(idx & 31);
  if (e >= E) return;
  int s = (int)ei[e];
  int d = (int)ei[E + e];
  long off = (long)d * HID + c;
  float m = fmaxf(h[(long)s * HID + c], 0.f) + 1e-7f;
  float ex = __expf(m - mx[off]);
  float alpha = ex / (den[off] + 1e-16f);
  atomicAdd(agg + off, alpha * m);
}

// ---------------------------------------------------------------- GEMM phase
// Generic C = A@B + bias.  One wave -> one 16x16 tile of C.
__global__ void k_gemm_bias(const float* __restrict__ A,
                            const float* __restrict__ B,
                            const float* __restrict__ bias,
                            float* __restrict__ C, int M, int K, int NC) {
  int lane = threadIdx.x & 31;
  int wib = threadIdx.x >> 5;
  int ntiles = NC >> 4;
  int w = blockIdx.x * (blockDim.x >> 5) + wib;
  int mt = w / ntiles, nt = w % ntiles;
  if (mt * 16 >= M) return;  // wave-uniform: whole idle waves exit
  int half = lane >> 4, l16 = lane & 15, koff = half * 2;
  int arow = mt * 16 + l16;
  if (arow >= M) arow = M - 1;  // data clamp keeps EXEC all-ones for WMMA
  int col = nt * 16 + l16;
  long abase = (long)arow * K;
  v8f acc = {};
  for (int kb = 0; kb < K; kb += 4) {
    v2f a, b;
    const float* ap = A + abase + kb + koff;
    a.x = ap[0];
    a.y = ap[1];
    const float* bp = B + (long)(kb + koff) * NC + col;
    b.x = bp[0];
    b.y = bp[NC];
    acc = wmma4(a, b, acc);
  }
  float bv = bias[col];
#pragma unroll
  for (int r = 0; r < 8; ++r) {
    int row = mt * 16 + r + half * 8;
    if (row < M) C[(long)row * NC + col] = acc[r] + bv;
  }
}

// z = (h + agg) @ w1 + b1 ; accumulate per-channel sum / sum-of-squares for BN
__global__ void k_gemm1(const float* __restrict__ h, const float* __restrict__ agg,
                        const float* __restrict__ w1, const float* __restrict__ b1,
                        float* __restrict__ z, float* __restrict__ bnsum,
                        float* __restrict__ bnsq, int M) {
  __shared__ float sSum[HID2], sSq[HID2];
  for (int i = threadIdx.x; i < HID2; i += blockDim.x) {
    sSum[i] = 0.f;
    sSq[i] = 0.f;
  }
  __syncthreads();
  int lane = threadIdx.x & 31;
  int wib = threadIdx.x >> 5;
  int w = blockIdx.x * (blockDim.x >> 5) + wib;
  int mt = w >> 2, nt = w & 3;  // NC = 64 -> 4 column tiles
  int half = lane >> 4, l16 = lane & 15, koff = half * 2;
  int col = nt * 16 + l16;
  bool active = (mt * 16 < M);
  if (active) {
    int arow = mt * 16 + l16;
    if (arow >= M) arow = M - 1;
    long abase = (long)arow * HID;
    v8f acc = {};
    for (int kb = 0; kb < HID; kb += 4) {
      v2f a, b;
      long ai = abase + kb + koff;
      a.x = h[ai] + agg[ai];        // fused residual
      a.y = h[ai + 1] + agg[ai + 1];
      const float* bp = w1 + (long)(kb + koff) * HID2 + col;
      b.x = bp[0];
      b.y = bp[HID2];
      acc = wmma4(a, b, acc);
    }
    float bv = b1[col];
    float ls = 0.f, lq = 0.f;
#pragma unroll
    for (int r = 0; r < 8; ++r) {
      int row = mt * 16 + r + half * 8;
      float zv = acc[r] + bv;
      if (row < M) {
        z[(long)row * HID2 + col] = zv;
        ls += zv;
        lq += zv * zv;
      }
    }
    atomicAdd(&sSum[col], ls);  // LDS float atomics (ds_add_f32)
    atomicAdd(&sSq[col], lq);
  }
  __syncthreads();
  for (int i = threadIdx.x; i < HID2; i += blockDim.x) {
    atomicAdd(&bnsum[i], sSum[i]);
    atomicAdd(&bnsq[i], sSq[i]);
  }
}

__global__ void k_bnfinal(const float* __restrict__ bnsum,
                          const float* __restrict__ bnsq,
                          const float* __restrict__ gamma,
                          const float* __restrict__ beta,
                          float* __restrict__ scale, float* __restrict__ shift,
                          int M) {
  int c = threadIdx.x;
  if (c < HID2) {
    float inv = 1.f / (float)M;
    float mean = bnsum[c] * inv;
    float var = bnsq[c] * inv - mean * mean;  // biased, as in BN train
    float rstd = rsqrtf(var + 1e-5f);
    float sc = gamma[c] * rstd;
    scale[c] = sc;
    shift[c] = beta[c] - mean * sc;
  }
}

// h_out = relu( relu(scale*z + shift) @ w2 + b2 )
__global__ void k_gemm2(const float* __restrict__ z, const float* __restrict__ scale,
                        const float* __restrict__ shift,
                        const float* __restrict__ w2, const float* __restrict__ b2,
                        float* __restrict__ hout, int M) {
  int lane = threadIdx.x & 31;
  int wib = threadIdx.x >> 5;
  int w = blockIdx.x * (blockDim.x >> 5) + wib;
  int mt = w >> 1, nt = w & 1;  // NC = 32 -> 2 column tiles
  if (mt * 16 >= M) return;
  int half = lane >> 4, l16 = lane & 15, koff = half * 2;
  int arow = mt * 16 + l16;
  if (arow >= M) arow = M - 1;
  int col = nt * 16 + l16;
  long abase = (long)arow * HID2;
  v8f acc = {};
  for (int kb = 0; kb < HID2; kb += 4) {
    int k0 = kb + koff, k1 = k0 + 1;
    v2f a, b;
    a.x = fmaxf(z[abase + k0] * scale[k0] + shift[k0], 0.f);  // fused BN+ReLU
    a.y = fmaxf(z[abase + k1] * scale[k1] + shift[k1], 0.f);
    const float* bp = w2 + (long)k0 * HID + col;
    b.x = bp[0];
    b.y = bp[HID];
    acc = wmma4(a, b, acc);
  }
  float bv = b2[col];
#pragma unroll
  for (int r = 0; r < 8; ++r) {
    int row = mt * 16 + r + half * 8;
    if (row < M) hout[(long)row * HID + col] = fmaxf(acc[r] + bv, 0.f);
  }
}

// ---------------------------------------------------------------- launcher
static inline int cdiv_l(long a, long b) { return (int)((a + b - 1) / b); }

extern "C" void kernel_launch(void* const* d_in, const int* in_sizes, int n_in,
                              void* d_out, int out_size, void* d_ws, size_t ws_size,
                              hipStream_t stream) {
  const float* x = (const float*)d_in[0];
  const long long* ei = (const long long*)d_in[1];  // int64 per reference
  const float* w0 = (const float*)d_in[2];
  const float* b0 = (const float*)d_in[3];
  const float* lin1_w = (const float*)d_in[4];  // [16][32][64]
  const float* lin1_b = (const float*)d_in[5];  // [16][64]
  const float* gamma = (const float*)d_in[6];   // [16][64]
  const float* beta = (const float*)d_in[7];    // [16][64]
  const float* lin2_w = (const float*)d_in[8];  // [16][64][32]
  const float* lin2_b = (const float*)d_in[9];  // [16][32]
  const float* w16 = (const float*)d_in[10];
  const float* b16 = (const float*)d_in[11];

  const int M = in_sizes[0] / 128;       // nodes
  const long E = (long)in_sizes[1] / 2;  // edges

  // workspace layout (floats): hA | hB | mx | den | agg | stats
  // z (M x 64) aliases the mx+den region (dead by the time GEMM1 writes z)
  float* ws = (float*)d_ws;
  const long NH = (long)M * HID;
  float* hA = ws;
  float* hB = hA + NH;
  float* mx = hB + NH;
  float* den = mx + NH;
  float* agg = den + NH;
  float* z = mx;
  float* stats = agg + NH;
  float* bnsum = stats;
  float* bnsq = stats + HID2;
  float* scale = stats + 2 * HID2;
  float* shift = stats + 3 * HID2;

  const int mtiles = cdiv_l(M, 16);
  const int eblocks = cdiv_l(E * 32, 256);

  // input projection: h = x @ w0 + b0   (K=128, NC=32)
  k_gemm_bias<<<cdiv_l((long)mtiles * 2, 8), 256, 0, stream>>>(x, w0, b0, hA, M,
                                                               128, HID);

  float* hc = hA;
  float* ho = hB;
  for (int l = 0; l < 16; ++l) {
    k_zero<<<2048, 256, 0, stream>>>(mx, 3 * NH);         // mx, den, agg
    k_zero<<<1, 256, 0, stream>>>(stats, 2 * HID2);       // bn sums
    k_edge_max<<<eblocks, 256, 0, stream>>>(hc, ei, mx, E);
    k_edge_expsum<<<eblocks, 256, 0, stream>>>(hc, ei, mx, den, E);
    k_edge_agg<<<eblocks, 256, 0, stream>>>(hc, ei, mx, den, agg, E);
    k_gemm1<<<cdiv_l((long)mtiles * 4, 8), 256, 0, stream>>>(
        hc, agg, lin1_w + (long)l * HID * HID2, lin1_b + l * HID2, z, bnsum,
        bnsq, M);
    k_bnfinal<<<1, 64, 0, stream>>>(bnsum, bnsq, gamma + l * HID2,
                                    beta + l * HID2, scale, shift, M);
    k_gemm2<<<cdiv_l((long)mtiles * 2, 8), 256, 0, stream>>>(
        z, scale, shift, lin2_w + (long)l * HID2 * HID, lin2_b + l * HID, ho, M);
    float* t = hc;
    hc = ho;
    ho = t;
  }

  // output projection: out = h @ w16 + b16   (K=32, NC=64)
  k_gemm_bias<<<cdiv_l((long)mtiles * 4, 8), 256, 0, stream>>>(
      hc, w16, b16, (float*)d_out, M, HID, HID2);
}